// TAB_Layer_88553635709491
// MI455X (gfx1250) — compile-verified
//
#include <hip/hip_runtime.h>

typedef __attribute__((ext_vector_type(2))) float v2f;
typedef __attribute__((ext_vector_type(4))) float v4f;
typedef __attribute__((ext_vector_type(8))) float v8f;

#define B_DIM 4096
#define T_DIM 16
#define F_DIM 512
#define TF (T_DIM * F_DIM)          // 8192
#define NCHUNK 8
#define B_PER_CHUNK (B_DIM / NCHUNK) // 512

// ---------------------------------------------------------------------------
// Stage 1: per-chunk column sums and sums-of-squares over B via WMMA.
//   D = A(16x4 of x) * B(4x16 ones) + C  ->  D[m,n] = rowsum_m (all n equal).
//   B = ones  => B-operand VGPR layout is irrelevant (layout-safe).
//   A layout (ISA 7.12.2, 32-bit A 16x4): lane m   -> K=0 (v0), K=1 (v1)
//                                         lane m+16-> K=2 (v0), K=3 (v1)
//   D layout: lane 0 -> M=0..7 in c[0..7]; lane 16 -> M=8..15.
// One wave (32 threads) handles a 16-wide tf tile over 512 b values.
// Default (RT) load hints keep x resident in the 192 MB L2 for stage 3.
// ---------------------------------------------------------------------------
__global__ __launch_bounds__(32) void stage1_wmma(const float* __restrict__ x,
                                                  float* __restrict__ psum,
                                                  float* __restrict__ psq) {
    const int tile  = blockIdx.x;          // 0 .. TF/16 - 1
    const int chunk = blockIdx.y;          // 0 .. NCHUNK - 1
    const int lane  = threadIdx.x;         // 0 .. 31
    const int m     = lane & 15;
    const int kb    = (lane >> 4) * 2;     // 0 or 2
    const int tf0   = tile * 16;

    const float* base =
        x + (size_t)(chunk * B_PER_CHUNK + kb) * TF + tf0 + m;

    v2f ones; ones.x = 1.0f; ones.y = 1.0f;
    v8f c0 = {}, c1 = {};   // sum accumulators
    v8f q0 = {}, q1 = {};   // sum-of-squares accumulators

    #pragma unroll 4
    for (int s = 0; s < B_PER_CHUNK / 4; s += 2) {
        const float* p0 = base + (size_t)(s * 4) * TF;
        const float* p1 = p0 + (size_t)4 * TF;
        v2f a0; a0.x = p0[0]; a0.y = p0[TF];
        v2f a1; a1.x = p1[0]; a1.y = p1[TF];
        v2f s0; s0.x = a0.x * a0.x; s0.y = a0.y * a0.y;
        v2f s1; s1.x = a1.x * a1.x; s1.y = a1.y * a1.y;
        c0 = __builtin_amdgcn_wmma_f32_16x16x4_f32(false, a0, false, ones,
                                                   (short)0, c0, false, false);
        q0 = __builtin_amdgcn_wmma_f32_16x16x4_f32(false, s0, false, ones,
                                                   (short)0, q0, false, false);
        c1 = __builtin_amdgcn_wmma_f32_16x16x4_f32(false, a1, false, ones,
                                                   (short)0, c1, false, false);
        q1 = __builtin_amdgcn_wmma_f32_16x16x4_f32(false, s1, false, ones,
                                                   (short)0, q1, false, false);
    }

    v8f cs = c0 + c1;
    v8f qs = q0 + q1;

    // lane 0 holds M=0..7, lane 16 holds M=8..15 (any N column; all equal).
    if (m == 0 && (lane == 0 || lane == 16)) {
        const int mbase = (lane >> 4) * 8;
        float* ds = psum + (size_t)chunk * TF + tf0 + mbase;
        float* dq = psq  + (size_t)chunk * TF + tf0 + mbase;
        #pragma unroll
        for (int i = 0; i < 8; ++i) { ds[i] = cs[i]; dq[i] = qs[i]; }
    }
}

// ---------------------------------------------------------------------------
// Stage 2: fold chunk partials -> prefix stats over t -> scale/shift tables.
//   scale[t,f] = inv_std * gamma * p[t]^2
//   shift[t,f] = (beta - mean * inv_std * gamma) * p[t]^2
// 8 blocks x 64 threads; thread f walks t keeping running prefix sums.
// ---------------------------------------------------------------------------
__global__ __launch_bounds__(64) void stage2(const float* __restrict__ psum,
                                             const float* __restrict__ psq,
                                             const float* __restrict__ p,
                                             const float* __restrict__ gamma,
                                             const float* __restrict__ beta,
                                             float* __restrict__ scale,
                                             float* __restrict__ shift) {
    const int f = blockIdx.x * 64 + threadIdx.x;   // 0 .. F_DIM-1
    float s = 0.0f, ss = 0.0f;
    #pragma unroll
    for (int t = 0; t < T_DIM; ++t) {
        const int tf = t * F_DIM + f;
        #pragma unroll
        for (int ch = 0; ch < NCHUNK; ++ch) {
            s  += psum[(size_t)ch * TF + tf];
            ss += psq [(size_t)ch * TF + tf];
        }
        const float cnt  = (float)((t + 1) * B_DIM);
        const float mean = s / cnt;
        const float var  = ss / cnt - mean * mean;
        const float inv  = rsqrtf(var + 1e-5f);
        const float pv   = p[t];
        const float p2   = pv * pv;
        const float g    = gamma[tf] * inv;
        scale[tf] = g * p2;
        shift[tf] = (beta[tf] - mean * g) * p2;
    }
}

// ---------------------------------------------------------------------------
// Stage 3: streaming y = fma(x, scale[tf], shift[tf]) with 128-bit accesses.
// x loads use default RT hints (hits L2 lines populated by stage 1);
// output stores are NON-TEMPORAL so the never-re-read result does not evict
// x from the 192 MB L2 (helps this call's reads and the next replay's
// stage-1 pass). scale/shift (64 KB) stay hot in cache.
// ---------------------------------------------------------------------------
__global__ __launch_bounds__(256) void stage3(const float* __restrict__ x,
                                              const float* __restrict__ scale,
                                              const float* __restrict__ shift,
                                              float* __restrict__ out) {
    const size_t gid  = (size_t)blockIdx.x * blockDim.x + threadIdx.x;
    const size_t base = gid * 4;
    const int tf4 = (int)(base & (size_t)(TF - 1)) >> 2;

    v4f xv = ((const v4f*)x)[gid];
    v4f sc = ((const v4f*)scale)[tf4];
    v4f sh = ((const v4f*)shift)[tf4];
    v4f y;
    y.x = fmaf(xv.x, sc.x, sh.x);
    y.y = fmaf(xv.y, sc.y, sh.y);
    y.z = fmaf(xv.z, sc.z, sh.z);
    y.w = fmaf(xv.w, sc.w, sh.w);
    __builtin_nontemporal_store(y, (v4f*)out + gid);
}

extern "C" void kernel_launch(void* const* d_in, const int* in_sizes, int n_in,
                              void* d_out, int out_size, void* d_ws, size_t ws_size,
                              hipStream_t stream) {
    const float* x     = (const float*)d_in[0];
    const float* p     = (const float*)d_in[1];
    const float* gamma = (const float*)d_in[2];
    const float* beta  = (const float*)d_in[3];
    float* out = (float*)d_out;

    float* psum  = (float*)d_ws;             // NCHUNK * TF
    float* psq   = psum + (size_t)NCHUNK * TF;
    float* scale = psq  + (size_t)NCHUNK * TF;
    float* shift = scale + TF;

    dim3 g1(TF / 16, NCHUNK);
    stage1_wmma<<<g1, 32, 0, stream>>>(x, psum, psq);
    stage2<<<F_DIM / 64, 64, 0, stream>>>(psum, psq, p, gamma, beta, scale, shift);

    const size_t n4 = (size_t)B_DIM * TF / 4;   // float4 count
    stage3<<<(unsigned)((n4 + 255) / 256), 256, 0, stream>>>(x, scale, shift, out);
}